// distribution_3393024163971
// MI455X (gfx1250) — compile-verified
//
#include <hip/hip_runtime.h>
#include <stdint.h>

#define B_   16
#define N_   2048
#define M_   2048
#define NP1  2049
#define MP1  2049
#define GAMMA_ 0.5f

#define THREADS 256
#define WAVES 8
#define ROWS_PER_WAVE 8
#define ROWS_PER_WG (WAVES * ROWS_PER_WAVE)              /* 64 */
#define RBLKS ((NP1 + ROWS_PER_WG - 1) / ROWS_PER_WG)    /* 33 */

typedef unsigned int u32;
typedef __attribute__((ext_vector_type(4))) u32 u32x4;
typedef __attribute__((ext_vector_type(8))) u32 u32x8;

// -------- CDNA5 Tensor Data Mover: 1-D row (nelem f32/i32) global -> LDS ----
// D# per CDNA5 ISA ch.8: group0 {count=1, lds_addr, global_addr, type=2},
// group1 {data_size=4B, tensor_dim0=nelem, tensor_dim1=1, tile_dim0=nelem,
// tile_dim1=tile_dim2=1, tensor_dim0_stride=nelem}. 2-group form (VADDR2/3=NULL).
__device__ __forceinline__ void tdm_load_1d_to_lds(void* lds_dst, const void* gsrc,
                                                   unsigned nelem) {
  // generic pointer to LDS: low 32 bits are the wave-relative LDS byte offset
  u32 lds_off = (u32)(uintptr_t)lds_dst;
  unsigned long long ga = (unsigned long long)(uintptr_t)gsrc;
  u32x4 g0 = { 1u,                                   // count=1, user mode
               lds_off,                              // lds_addr (bytes)
               (u32)ga,                              // global_addr[31:0]
               ((u32)(ga >> 32) & 0x01FFFFFFu) | (2u << 30) }; // addr[56:32] | type=2
  u32x8 g1;
  g1[0] = (2u << 16);                    // wg_mask=0, data_size=2 (4 bytes)
  g1[1] = (nelem & 0xFFFFu) << 16;       // tensor_dim0[15:0]
  g1[2] = (nelem >> 16) | (1u << 16);    // tensor_dim0[31:16] | tensor_dim1=1
  g1[3] = (nelem & 0xFFFFu) << 16;       // tensor_dim1 hi=0   | tile_dim0
  g1[4] = 0x00010001u;                   // tile_dim1=1, tile_dim2=1
  g1[5] = nelem;                         // tensor_dim0_stride[31:0]
  g1[6] = (nelem & 0xFFFFu) << 16;       // stride0 hi=0 | tensor_dim1_stride lo
  g1[7] = 0u;
  asm volatile("tensor_load_to_lds %0, %1" :: "s"(g0), "s"(g1) : "memory");
}

// ---------------- kernel 1: gather pos0[b,i], pos1[b,j] ---------------------
__global__ void gather_pos_kernel(const float* __restrict__ scores,
                                  const int* __restrict__ gt0,
                                  const int* __restrict__ gt1,
                                  float* __restrict__ pos0,
                                  float* __restrict__ pos1) {
  int t = blockIdx.x * blockDim.x + threadIdx.x;
  if (t < B_ * N_) {
    int b = t / N_, i = t % N_;
    int g = gt0[t]; if (g < 0) g = M_;
    pos0[t] = scores[((size_t)b * NP1 + i) * MP1 + g];
  } else if (t < B_ * N_ + B_ * M_) {
    int u = t - B_ * N_;
    int b = u / M_, j = u % M_;
    int g = gt1[u]; if (g < 0) g = N_;
    pos1[u] = scores[((size_t)b * NP1 + g) * MP1 + j];
  }
}

// ---------------- kernel 2: fused single pass over scores ------------------
__global__ void __launch_bounds__(THREADS)
fused_pass_kernel(const float* __restrict__ scores,
                  const int* __restrict__ gt0g,
                  const int* __restrict__ gt1g,
                  const float* __restrict__ pos0g,
                  const float* __restrict__ pos1g,
                  float* __restrict__ colsum,
                  float* __restrict__ loss1,
                  double* __restrict__ dacc) {
  __shared__ float s_pos1[M_];
  __shared__ int   s_gt1[M_];
  __shared__ float s_col[M_];
  __shared__ float s_red[2 * WAVES];

  const int b    = blockIdx.x / RBLKS;
  const int rb   = blockIdx.x % RBLKS;
  const int row0 = rb * ROWS_PER_WG;
  const int tid  = threadIdx.x;
  const int wave = tid >> 5;
  const int lane = tid & 31;

  for (int j = tid; j < M_; j += THREADS) s_col[j] = 0.0f;

  if (wave == 0) {
    // CDNA5 TDM: stage this batch's pos1 + gt1 rows into LDS once per block
    tdm_load_1d_to_lds(s_pos1, pos1g + (size_t)b * M_, M_);
    tdm_load_1d_to_lds(s_gt1,  gt1g  + (size_t)b * M_, M_);
    __builtin_amdgcn_s_wait_tensorcnt(0);
  }
  __syncthreads();

  // one-time remap of gt1 (-1 -> N) so the hot loop is compare-only
  for (int j = tid; j < M_; j += THREADS) {
    int v = s_gt1[j]; if (v < 0) s_gt1[j] = N_;
  }
  __syncthreads();

  const float* plane = scores + (size_t)b * NP1 * MP1;

  // per-wave row block state
  int          iv[ROWS_PER_WAVE];
  bool         rowv[ROWS_PER_WAVE], colv[ROWS_PER_WAVE];
  int          g0v[ROWS_PER_WAVE];
  float        p0v[ROWS_PER_WAVE];
  const float* rowp[ROWS_PER_WAVE];
  float        rsum[ROWS_PER_WAVE];

  const int ibase = row0 + wave * ROWS_PER_WAVE;
#pragma unroll
  for (int r = 0; r < ROWS_PER_WAVE; ++r) {
    int i = ibase + r;
    iv[r]   = i;
    colv[r] = (i < NP1);
    rowv[r] = (i < N_);
    int ic  = colv[r] ? i : (NP1 - 1);          // clamp for addressing only
    rowp[r] = plane + (size_t)ic * MP1;
    int g0 = 0; float p0 = 0.0f;
    if (rowv[r]) {
      g0 = gt0g[b * N_ + i]; if (g0 < 0) g0 = M_;
      p0 = pos0g[b * N_ + i];
    }
    g0v[r] = g0; p0v[r] = p0;
    rsum[r] = 0.0f;
  }

  float vs = 0.0f, vs2 = 0.0f;

  // column-step outer loop: LDS reads + the s_col atomic are amortized
  // over all ROWS_PER_WAVE rows; 8 independent NT loads per step for MLP.
  for (int j = lane; j < MP1; j += 32) {
    float sv[ROWS_PER_WAVE];
#pragma unroll
    for (int r = 0; r < ROWS_PER_WAVE; ++r)
      sv[r] = __builtin_nontemporal_load(rowp[r] + j);

    const bool jcol = (j < M_);
    int   g1v = 0;
    float p1  = 0.0f;
    if (jcol) { g1v = s_gt1[j]; p1 = s_pos1[j]; }

    float colAcc = 0.0f;
#pragma unroll
    for (int r = 0; r < ROWS_PER_WAVE; ++r) {
      float s = sv[r];
      if (rowv[r] && j != g0v[r]) {             // row-margin term
        float m1 = s - p0v[r];
        rsum[r] += fmaxf(m1 + GAMMA_, 0.0f);
        vs += m1; vs2 = fmaf(m1, m1, vs2);
      }
      if (jcol && colv[r] && iv[r] != g1v) {    // col-margin term
        float m2 = s - p1;
        colAcc += fmaxf(m2 + GAMMA_, 0.0f);
        vs += m2; vs2 = fmaf(m2, m2, vs2);
      }
    }
    if (jcol) atomicAdd(&s_col[j], colAcc);     // one ds_add_f32 per 8 rows
  }

#pragma unroll
  for (int r = 0; r < ROWS_PER_WAVE; ++r) {
    if (rowv[r]) {
      float rs = rsum[r];
      for (int m = 16; m > 0; m >>= 1) rs += __shfl_xor(rs, m, 32);
      if (lane == 0) atomicAdd(&loss1[b], __logf(rs + 1.0f));
    }
  }

  __syncthreads();
  for (int j = tid; j < M_; j += THREADS)
    atomicAdd(&colsum[(size_t)b * M_ + j], s_col[j]);

  for (int m = 16; m > 0; m >>= 1) {
    vs  += __shfl_xor(vs,  m, 32);
    vs2 += __shfl_xor(vs2, m, 32);
  }
  if (lane == 0) { s_red[wave] = vs; s_red[WAVES + wave] = vs2; }
  __syncthreads();
  if (tid == 0) {
    double a = 0.0, c = 0.0;
    for (int w = 0; w < WAVES; ++w) {
      a += (double)s_red[w];
      c += (double)s_red[WAVES + w];
    }
    atomicAdd(&dacc[0], a);
    atomicAdd(&dacc[1], c);
  }
}

// ---------------- kernel 3: finalize per-batch outputs ---------------------
__global__ void finalize_kernel(const float* __restrict__ colsum,
                                const float* __restrict__ loss1,
                                const double* __restrict__ dacc,
                                float* __restrict__ out) {
  __shared__ float sred[THREADS];
  int b = blockIdx.x;
  float acc = 0.0f;
  for (int j = threadIdx.x; j < M_; j += THREADS)
    acc += __logf(colsum[(size_t)b * M_ + j] + 1.0f);
  sred[threadIdx.x] = acc;
  __syncthreads();
  for (int s = THREADS / 2; s > 0; s >>= 1) {
    if (threadIdx.x < s) sred[threadIdx.x] += sred[threadIdx.x + s];
    __syncthreads();
  }
  if (threadIdx.x == 0) {
    double cnt = 2.0 * (double)B_ * (double)N_ * (double)M_;
    double vsum = dacc[0], vsum2 = dacc[1];
    double var = (vsum2 - vsum * vsum / cnt) / (cnt - 1.0);
    float var_loss = (float)log(var + 1.0);
    float lm1 = loss1[b] / (float)N_;
    float lm2 = sred[0] / (float)M_;
    out[b] = 0.5f * (lm1 + lm2) + var_loss;
  }
}

extern "C" void kernel_launch(void* const* d_in, const int* in_sizes, int n_in,
                              void* d_out, int out_size, void* d_ws, size_t ws_size,
                              hipStream_t stream) {
  (void)in_sizes; (void)n_in; (void)out_size; (void)ws_size;
  const float* scores = (const float*)d_in[0];
  const int*   gt0    = (const int*)d_in[1];
  const int*   gt1    = (const int*)d_in[2];
  float* out = (float*)d_out;

  // workspace layout (doubles first for alignment)
  double* dacc  = (double*)d_ws;            // 2 doubles: sum(m), sum(m^2)
  float*  loss1 = (float*)d_ws + 4;         // 16 floats: sum_i log(rowsum+1) per b
  float*  pos0  = (float*)d_ws + 20;        // B*N
  float*  pos1  = pos0 + B_ * N_;           // B*M
  float*  colsum = pos1 + B_ * M_;          // B*M
  size_t used_bytes = (size_t)(20 + 3 * B_ * M_) * sizeof(float);

  hipMemsetAsync(d_ws, 0, used_bytes, stream);

  int gth = B_ * N_ + B_ * M_;
  gather_pos_kernel<<<(gth + 255) / 256, 256, 0, stream>>>(scores, gt0, gt1, pos0, pos1);
  fused_pass_kernel<<<B_ * RBLKS, THREADS, 0, stream>>>(scores, gt0, gt1, pos0, pos1,
                                                        colsum, loss1, dacc);
  finalize_kernel<<<B_, THREADS, 0, stream>>>(colsum, loss1, dacc, out);
}